// FCOSDecoder_44676249813201
// MI455X (gfx1250) — compile-verified
//
#include <hip/hip_runtime.h>
#include <stdint.h>
#include <stddef.h>

#define NUM_CLASSES 80
#define BATCH 16
#define TOPN 1000
#define NCAND 3264
#define MAXDET 100
#define TOTPOS 21824   // 16384+4096+1024+256+64

struct Heads {
    const float* cls[5];
    const float* reg[5];
    const float* ctr[5];
};

__constant__ int c_HW[5]     = {16384, 4096, 1024, 256, 64};
__constant__ int c_LOFF[5]   = {0, 16384, 20480, 21504, 21760};
__constant__ int c_STRIDE[5] = {8, 16, 32, 64, 128};
__constant__ int c_WD[5]     = {128, 64, 32, 16, 8};
__constant__ int c_COFF[5]   = {0, 1000, 2000, 3000, 3256};
__constant__ int c_K[5]      = {1000, 1000, 1000, 256, 64};

__device__ __forceinline__ float sigm(float x) { return 1.0f / (1.0f + expf(-x)); }

// ---------------------------------------------------------------------------
// Stage 1: per-position score + argmax class.
// 682 blocks x 8 tiles; tile = 64 positions (64*80*4B = 20480B cls data).
// cls tile is DMA'd global->LDS with global_load_async_to_lds_b128,
// double-buffered across tiles; per-tile wait is s_wait_asynccnt <= 5.
// Each wave32 owns one position at a time: lanes own channels, butterfly
// argmax reduction across the wave.
// ---------------------------------------------------------------------------
__global__ __launch_bounds__(256) void stage1_scores(Heads h, float* __restrict__ scores,
                                                     int* __restrict__ classes)
{
    __shared__ __align__(16) float stage[2][64 * NUM_CLASSES];  // 2 x 20KB
    const int tid  = threadIdx.x;
    const int wave = tid >> 5;
    const int lane = tid & 31;
    const int tilesPerBatch = 341;  // 256+64+16+4+1

    auto tileInfo = [&](int t, int& b, int& lvl, int& pos0) {
        b = t / tilesPerBatch;
        int r = t - b * tilesPerBatch;
        if      (r < 256) { lvl = 0; pos0 = r * 64; }
        else if (r < 320) { lvl = 1; pos0 = (r - 256) * 64; }
        else if (r < 336) { lvl = 2; pos0 = (r - 320) * 64; }
        else if (r < 340) { lvl = 3; pos0 = (r - 336) * 64; }
        else              { lvl = 4; pos0 = 0; }
    };

    auto issue = [&](int t, int bufsel) {
        int b, lvl, pos0; tileInfo(t, b, lvl, pos0);
        const float* gp = h.cls[lvl] + ((size_t)b * c_HW[lvl] + pos0) * NUM_CLASSES;
        unsigned ldsbase = (unsigned)(size_t)(void*)(&stage[bufsel][0]);  // low 32b of flat = LDS offset
#pragma unroll
        for (int i = 0; i < 5; i++) {
            unsigned off   = (unsigned)(i * 4096 + tid * 16);  // 5*256*16B = 20480B
            unsigned laddr = ldsbase + off;
            asm volatile("global_load_async_to_lds_b128 %0, %1, %2"
                         :: "v"(laddr), "v"(off), "s"(gp) : "memory");
        }
    };

    const int t0 = blockIdx.x * 8;
    issue(t0, 0);
    for (int k = 0; k < 8; k++) {
        const int t = t0 + k;
        if (k + 1 < 8) {
            issue(t + 1, (k + 1) & 1);                       // prefetch next tile
            asm volatile("s_wait_asynccnt 0x5" ::: "memory"); // tile k complete (in-order)
        } else {
            asm volatile("s_wait_asynccnt 0x0" ::: "memory");
        }
        __syncthreads();

        int b, lvl, pos0; tileInfo(t, b, lvl, pos0);
        const float* buf = &stage[k & 1][0];
#pragma unroll
        for (int pp = 0; pp < 8; pp++) {
            const int p = wave * 8 + pp;
            const float* row = buf + p * NUM_CLASSES;
            float v0 = row[lane];
            float v1 = row[lane + 32];
            float v2 = (lane < 16) ? row[lane + 64] : -3.4e38f;
            float v = v0; int ci = lane;
            if (v1 > v) { v = v1; ci = lane + 32; }
            if (v2 > v) { v = v2; ci = lane + 64; }
#pragma unroll
            for (int off = 16; off > 0; off >>= 1) {         // wave32 butterfly argmax
                float ov = __shfl_xor(v, off, 32);
                int   oi = __shfl_xor(ci, off, 32);
                if (ov > v || (ov == v && oi < ci)) { v = ov; ci = oi; }
            }
            if (lane == 0) {
                const int gpos = pos0 + p;
                const float ctr = h.ctr[lvl][(size_t)b * c_HW[lvl] + gpos];
                const float sc  = sqrtf(sigm(v) * sigm(ctr));
                const int widx  = b * TOTPOS + c_LOFF[lvl] + gpos;
                scores[widx]  = sc;
                classes[widx] = ci;
            }
        }
        __syncthreads();  // buffer k&1 may be re-issued next iteration
    }
}

// ---------------------------------------------------------------------------
// Stage 2: exact top-1000 threshold per (batch, level 0..2) via 4-round
// radix select on score bits (scores >= 0 -> IEEE bits are order-preserving).
// Emits threshold T and the count of ties-at-T to take.
// ---------------------------------------------------------------------------
__global__ __launch_bounds__(256) void select_topk(const float* __restrict__ scores,
                                                   float* __restrict__ thr, int* __restrict__ tie)
{
    const int b = blockIdx.x / 3, lvl = blockIdx.x % 3;
    const float* s = scores + b * TOTPOS + c_LOFF[lvl];
    const int n = c_HW[lvl];
    __shared__ unsigned hist[256];
    __shared__ unsigned s_prefix;
    __shared__ int s_rem;
    if (threadIdx.x == 0) { s_prefix = 0; s_rem = TOPN; }
    __syncthreads();
    for (int round = 0; round < 4; round++) {
        const int shift = 24 - 8 * round;
        hist[threadIdx.x] = 0;
        __syncthreads();
        const unsigned prefix = s_prefix;
        const unsigned hmask  = (round == 0) ? 0u : (0xFFFFFFFFu << (shift + 8));
        for (int i = threadIdx.x; i < n; i += 256) {
            const unsigned u = __float_as_uint(s[i]);
            if ((u & hmask) == (prefix & hmask))
                atomicAdd(&hist[(u >> shift) & 255], 1u);
        }
        __syncthreads();
        if (threadIdx.x == 0) {
            int rem = s_rem; unsigned acc = 0;
            for (int v = 255; v >= 0; v--) {
                const unsigned c = hist[v];
                if (acc + c >= (unsigned)rem) {
                    s_prefix = prefix | ((unsigned)v << shift);
                    s_rem = rem - (int)acc;
                    break;
                }
                acc += c;
            }
        }
        __syncthreads();
    }
    if (threadIdx.x == 0) {
        thr[b * 5 + lvl] = __uint_as_float(s_prefix);
        tie[b * 5 + lvl] = s_rem;
    }
}

// ---------------------------------------------------------------------------
// Stage 3: compaction + box decode. One thread per position.
// ---------------------------------------------------------------------------
__global__ __launch_bounds__(256) void compact_decode(Heads h,
        const float* __restrict__ scores, const int* __restrict__ classes,
        const float* __restrict__ thr, const int* __restrict__ tie,
        int* __restrict__ gtc, int* __restrict__ eqc,
        float* __restrict__ candS, float* __restrict__ candC, float* __restrict__ candB)
{
    const int gid = blockIdx.x * 256 + threadIdx.x;
    if (gid >= BATCH * TOTPOS) return;
    const int b = gid / TOTPOS;
    const int r = gid - b * TOTPOS;
    int lvl, pos;
    if      (r < 16384) { lvl = 0; pos = r; }
    else if (r < 20480) { lvl = 1; pos = r - 16384; }
    else if (r < 21504) { lvl = 2; pos = r - 20480; }
    else if (r < 21760) { lvl = 3; pos = r - 21504; }
    else                { lvl = 4; pos = r - 21760; }

    const float s = scores[gid];
    int slot;
    if (lvl < 3) {
        const float T = thr[b * 5 + lvl];
        if (s > T) {
            slot = atomicAdd(&gtc[b * 5 + lvl], 1);          // exactly K - tie of these
        } else if (s == T) {
            const int tcap = tie[b * 5 + lvl];
            const int e = atomicAdd(&eqc[b * 5 + lvl], 1);
            if (e >= tcap) return;
            slot = (c_K[lvl] - tcap) + e;
        } else {
            return;
        }
    } else {
        slot = pos;
    }

    const int y = pos / c_WD[lvl], x = pos - y * c_WD[lvl];
    const float cx = (x + 0.5f) * (float)c_STRIDE[lvl];
    const float cy = (y + 0.5f) * (float)c_STRIDE[lvl];
    const float* rp = h.reg[lvl] + ((size_t)b * c_HW[lvl] + pos) * 4;
    const float e0 = expf(rp[0]), e1 = expf(rp[1]), e2 = expf(rp[2]), e3 = expf(rp[3]);
    float x1 = truncf(cx - e0), y1 = truncf(cy - e1);
    float x2 = truncf(cx + e2), y2 = truncf(cy + e3);
    x1 = fmaxf(x1, 0.0f);     y1 = fmaxf(y1, 0.0f);
    x2 = fminf(x2, 1023.0f);  y2 = fminf(y2, 1023.0f);

    const int ci = b * NCAND + c_COFF[lvl] + slot;
    candS[ci] = (s > 0.05f) ? s : -1.0f;
    candC[ci] = (float)classes[gid];
    float* bp = candB + (size_t)ci * 4;
    bp[0] = x1; bp[1] = y1; bp[2] = x2; bp[3] = y2;
}

// ---------------------------------------------------------------------------
// Stage 4: greedy NMS, one workgroup per batch, candidates in LDS (91KB).
// ---------------------------------------------------------------------------
__global__ __launch_bounds__(256) void nms_kernel(const float* __restrict__ candS,
                                                  const float* __restrict__ candC,
                                                  const float* __restrict__ candB,
                                                  float* __restrict__ out)
{
    const int b = blockIdx.x, tid = threadIdx.x;
    extern __shared__ float sm[];
    float* ss = sm;                // mutable scores
    float* sc = sm + NCAND;
    float* x1 = sm + 2 * NCAND;
    float* y1 = sm + 3 * NCAND;
    float* x2 = sm + 4 * NCAND;
    float* y2 = sm + 5 * NCAND;
    float* ar = sm + 6 * NCAND;
    __shared__ float rv[256];
    __shared__ int   ri[256];

    for (int i = tid; i < NCAND; i += 256) {
        ss[i] = candS[b * NCAND + i];
        sc[i] = candC[b * NCAND + i];
        const float* bp = candB + ((size_t)b * NCAND + i) * 4;
        const float a = bp[0], bb = bp[1], c = bp[2], d = bp[3];
        x1[i] = a; y1[i] = bb; x2[i] = c; y2[i] = d;
        ar[i] = (c - a) * (d - bb);
    }
    __syncthreads();

    float* outS = out;
    float* outC = out + BATCH * MAXDET;
    float* outB = out + 2 * BATCH * MAXDET;

    for (int it = 0; it < MAXDET; it++) {
        float bv = -3.4e38f; int bi = 0;
        for (int i = tid; i < NCAND; i += 256) {
            const float v = ss[i];
            if (v > bv || (v == bv && i < bi)) { bv = v; bi = i; }
        }
        rv[tid] = bv; ri[tid] = bi;
        __syncthreads();
        for (int w = 128; w > 0; w >>= 1) {
            if (tid < w) {
                const float ov = rv[tid + w]; const int oi = ri[tid + w];
                if (ov > rv[tid] || (ov == rv[tid] && oi < ri[tid])) { rv[tid] = ov; ri[tid] = oi; }
            }
            __syncthreads();
        }
        const int j = ri[0];
        const float best = rv[0];
        const bool valid = best > 0.0f;
        if (tid == 0) {
            outS[b * MAXDET + it] = valid ? best  : -1.0f;
            outC[b * MAXDET + it] = valid ? sc[j] : -1.0f;
            float* ob = outB + ((size_t)b * MAXDET + it) * 4;
            if (valid) { ob[0] = x1[j]; ob[1] = y1[j]; ob[2] = x2[j]; ob[3] = y2[j]; }
            else       { ob[0] = ob[1] = ob[2] = ob[3] = -1.0f; }
        }
        const float jx1 = x1[j], jy1 = y1[j], jx2 = x2[j], jy2 = y2[j], ja = ar[j];
        __syncthreads();
        for (int i = tid; i < NCAND; i += 256) {
            const float tlx = fmaxf(x1[i], jx1), tly = fmaxf(y1[i], jy1);
            const float brx = fminf(x2[i], jx2), bry = fminf(y2[i], jy2);
            const float w  = fmaxf(brx - tlx, 0.0f);
            const float hh = fmaxf(bry - tly, 0.0f);
            const float inter = w * hh;
            const float uni = fmaxf(ar[i] + ja - inter, 1e-4f);
            if (inter / uni >= 0.6f) ss[i] = -1.0f;
        }
        __syncthreads();
        if (tid == 0) ss[j] = -1.0f;
        __syncthreads();
    }
}

// ---------------------------------------------------------------------------
__global__ __launch_bounds__(256) void init_ws_kernel(int* gtc, int* eqc,
                                                      float* candS, float* candC, float* candB)
{
    const int gid = blockIdx.x * 256 + threadIdx.x;
    if (gid < 80) { gtc[gid] = 0; eqc[gid] = 0; }
    if (gid < BATCH * NCAND) {
        candS[gid] = -1.0f;
        candC[gid] = -1.0f;
        float* bp = candB + (size_t)gid * 4;
        bp[0] = bp[1] = bp[2] = bp[3] = -1.0f;
    }
}

extern "C" void kernel_launch(void* const* d_in, const int* in_sizes, int n_in,
                              void* d_out, int out_size, void* d_ws, size_t ws_size,
                              hipStream_t stream)
{
    (void)in_sizes; (void)n_in; (void)out_size; (void)ws_size;
    Heads h;
    for (int l = 0; l < 5; l++) {
        h.cls[l] = (const float*)d_in[3 * l + 0];
        h.reg[l] = (const float*)d_in[3 * l + 1];
        h.ctr[l] = (const float*)d_in[3 * l + 2];
    }
    char* ws = (char*)d_ws;
    float* scores  = (float*)ws; ws += sizeof(float) * BATCH * TOTPOS;
    int*   classes = (int*)ws;   ws += sizeof(int)   * BATCH * TOTPOS;
    float* thr     = (float*)ws; ws += sizeof(float) * 80;
    int*   tie     = (int*)ws;   ws += sizeof(int)   * 80;
    int*   gtc     = (int*)ws;   ws += sizeof(int)   * 80;
    int*   eqc     = (int*)ws;   ws += sizeof(int)   * 80;
    float* candS   = (float*)ws; ws += sizeof(float) * BATCH * NCAND;
    float* candC   = (float*)ws; ws += sizeof(float) * BATCH * NCAND;
    float* candB   = (float*)ws; ws += sizeof(float) * BATCH * NCAND * 4;

    init_ws_kernel<<<(BATCH * NCAND + 255) / 256, 256, 0, stream>>>(gtc, eqc, candS, candC, candB);
    stage1_scores<<<682, 256, 0, stream>>>(h, scores, classes);
    select_topk<<<BATCH * 3, 256, 0, stream>>>(scores, thr, tie);
    compact_decode<<<(BATCH * TOTPOS) / 256, 256, 0, stream>>>(h, scores, classes, thr, tie,
                                                               gtc, eqc, candS, candC, candB);
    nms_kernel<<<BATCH, 256, sizeof(float) * NCAND * 7, stream>>>(candS, candC, candB, (float*)d_out);
}